// BilateralFilter_54941221650477
// MI455X (gfx1250) — compile-verified
//
#include <hip/hip_runtime.h>

// Bilateral filter, D=9 (radius 4, circular mask -> 49 taps), sigma_color =
// sigma_space = 75, fp32 NHWC 8x1024x1024x3, reflect padding.
//
// MI455X strategy:
//  - LDS tiling (40x40x3 halo tile, 18.75 KB of the 320 KB/WGP) so HBM traffic
//    stays at the ~200 MB roofline floor (~8.6 us @ 23.3 TB/s).
//  - Data staged with CDNA5 async global->LDS loads (ASYNCcnt) +
//    s_wait_asynccnt, instead of load-to-VGPR-store-to-LDS.
//  - Exactly one v_exp_f32 per tap: w = exp2(K*diff^2 + K*r2),
//    K = -0.5/sigma^2 * log2(e); spatial term is a compile-time constant.
//  - WMMA intentionally not used: weights are data-dependent, no GEMM exists.

#define IMG_H 1024
#define IMG_W 1024
#define TILE  32
#define RAD   4
#define HT    (TILE + 2 * RAD)   // 40 halo pixels per side
#define LROW  (HT * 3)           // 120 floats per LDS row
#define NPIX  (HT * HT)          // 1600 halo pixels per tile
#define NTAPS 49

// -0.5 / (75*75) * log2(e): one constant serves both color and space terms.
static constexpr float KC = -0.5f / (75.0f * 75.0f) * 1.4426950408889634f;

typedef __attribute__((address_space(1))) int* gas_i32p;  // global AS
typedef __attribute__((address_space(3))) int* las_i32p;  // LDS AS

__device__ __forceinline__ void async_copy_b32(const float* g, float* l) {
#if __has_builtin(__builtin_amdgcn_global_load_async_to_lds_b32)
  __builtin_amdgcn_global_load_async_to_lds_b32(
      (gas_i32p)g, (las_i32p)l, /*imm offset*/ 0, /*cpol*/ 0);
#else
  // Generic LDS pointer keeps the LDS byte offset in its low 32 bits.
  unsigned lds_off = (unsigned)(size_t)l;
  asm volatile("global_load_async_to_lds_b32 %0, %1, off"
               :: "v"(lds_off), "v"(g)
               : "memory");
#endif
}

__device__ __forceinline__ void wait_async0() {
#if __has_builtin(__builtin_amdgcn_s_wait_asynccnt)
  __builtin_amdgcn_s_wait_asynccnt(0);
#else
  asm volatile("s_wait_asynccnt 0x0" ::: "memory");
#endif
}

__global__ __launch_bounds__(256) void
bilateral_9x9_kernel(const float* __restrict__ in, float* __restrict__ out) {
  __shared__ float tile[HT * LROW];  // 19200 B

  const int tid = (int)threadIdx.x;
  const int tx0 = (int)blockIdx.x * TILE - RAD;
  const int ty0 = (int)blockIdx.y * TILE - RAD;
  const size_t imgoff = (size_t)blockIdx.z * (size_t)IMG_H * IMG_W * 3;
  const float* img = in + imgoff;

  // ---- Stage halo tile into LDS with async global->LDS copies ----
  for (int p = tid; p < NPIX; p += 256) {
    const int row = p / HT;
    const int col = p - row * HT;
    int gy = ty0 + row;
    gy = (gy < 0) ? -gy : ((gy >= IMG_H) ? (2 * IMG_H - 2 - gy) : gy);
    int gx = tx0 + col;
    gx = (gx < 0) ? -gx : ((gx >= IMG_W) ? (2 * IMG_W - 2 - gx) : gx);
    const float* src = img + ((size_t)gy * IMG_W + gx) * 3;
    float* dst = &tile[row * LROW + col * 3];
    async_copy_b32(src + 0, dst + 0);
    async_copy_b32(src + 1, dst + 1);
    async_copy_b32(src + 2, dst + 2);
  }
  wait_async0();
  __syncthreads();

  // ---- 49-tap circular mask, fully unrolled at compile time ----
  static constexpr int DYT[NTAPS] = {
      -4,
      -3, -3, -3, -3, -3,
      -2, -2, -2, -2, -2, -2, -2,
      -1, -1, -1, -1, -1, -1, -1,
       0,  0,  0,  0,  0,  0,  0,  0,  0,
       1,  1,  1,  1,  1,  1,  1,
       2,  2,  2,  2,  2,  2,  2,
       3,  3,  3,  3,  3,
       4};
  static constexpr int DXT[NTAPS] = {
       0,
      -2, -1,  0,  1,  2,
      -3, -2, -1,  0,  1,  2,  3,
      -3, -2, -1,  0,  1,  2,  3,
      -4, -3, -2, -1,  0,  1,  2,  3,  4,
      -3, -2, -1,  0,  1,  2,  3,
      -3, -2, -1,  0,  1,  2,  3,
      -2, -1,  0,  1,  2,
       0};

#pragma unroll 1
  for (int k = 0; k < 4; ++k) {
    const int q  = tid + k * 256;        // 0..1023 within the 32x32 tile
    const int oy = q >> 5;
    const int ox = q & 31;

    const float* cen = &tile[(oy + RAD) * LROW + (ox + RAD) * 3];
    const float c0 = cen[0], c1 = cen[1], c2 = cen[2];

    float n0 = 0.0f, n1 = 0.0f, n2 = 0.0f, den = 0.0f;

#pragma unroll
    for (int t = 0; t < NTAPS; ++t) {
      const int dy = DYT[t];
      const int dx = DXT[t];
      const float* p = &tile[(oy + RAD + dy) * LROW + (ox + RAD + dx) * 3];
      const float a0 = p[0], a1 = p[1], a2 = p[2];
      const float d  = fabsf(a0 - c0) + fabsf(a1 - c1) + fabsf(a2 - c2);
      // exp(space_coeff*r2 + color_coeff*d^2) as a single v_exp_f32
      const float e = fmaf(KC, d * d, KC * (float)(dy * dy + dx * dx));
      const float w = __builtin_amdgcn_exp2f(e);
      n0 = fmaf(w, a0, n0);
      n1 = fmaf(w, a1, n1);
      n2 = fmaf(w, a2, n2);
      den += w;
    }

    const float inv = __builtin_amdgcn_rcpf(den);
    const size_t o = imgoff +
        (((size_t)((int)blockIdx.y * TILE + oy)) * IMG_W +
         ((int)blockIdx.x * TILE + ox)) * 3;
    out[o + 0] = n0 * inv;
    out[o + 1] = n1 * inv;
    out[o + 2] = n2 * inv;
  }
}

extern "C" void kernel_launch(void* const* d_in, const int* in_sizes, int n_in,
                              void* d_out, int out_size, void* d_ws, size_t ws_size,
                              hipStream_t stream) {
  const float* in = (const float*)d_in[0];
  float* out = (float*)d_out;
  const int B = in_sizes[0] / (IMG_H * IMG_W * 3);  // 8 for the reference shape

  dim3 grid(IMG_W / TILE, IMG_H / TILE, B);
  dim3 block(256, 1, 1);
  bilateral_9x9_kernel<<<grid, block, 0, stream>>>(in, out);

  (void)n_in; (void)out_size; (void)d_ws; (void)ws_size;
}